// RWKV_Time_Mixing_6176162972107
// MI455X (gfx1250) — compile-verified
//
#include <hip/hip_runtime.h>
#include <math.h>
#include <stdint.h>

#define CDIM 4096
#define KSPLIT 8
#define KCHUNK (CDIM / KSPLIT) // 512 columns per K-split
#define KC 64                  // columns per LDS tile
#define ROWS 64                // rows per workgroup
#define LDS_STRIDE 68          // padded row stride (floats): 4-bank shift per row

typedef float v2f __attribute__((ext_vector_type(2)));
typedef float v8f __attribute__((ext_vector_type(8)));

// ---------------------------------------------------------------------------
// Workspace layout (floats):
//   xk   : [0,      C)
//   xv   : [C,     2C)
//   xr   : [2C,    3C)
//   kkp  : [3C,   11C)   KSPLIT partials of kw@xk
//   vvp  : [11C,  19C)   KSPLIT partials of vw@xv
//   rrp  : [19C,  27C)   KSPLIT partials of rw@xr
//   rwkv : [27C,  28C)
//   outp : [28C,  36C)   KSPLIT partials of ow@rwkv
// ---------------------------------------------------------------------------

// gfx1250 async DMA: copies 16 bytes from the global address to this lane's LDS byte address.
// Tracked by ASYNCcnt. Generic shared pointers carry the LDS byte address in
// their low 32 bits (aperture check is on the high half), so truncation is
// the raw LDS address the instruction wants.
__device__ __forceinline__ void async_copy_b128(uint32_t lds_addr,
                                                const float* gptr) {
    asm volatile("global_load_async_to_lds_b128 %0, %1, off"
                 :
                 : "v"(lds_addr), "v"((uint64_t)(uintptr_t)gptr)
                 : "memory");
}
__device__ __forceinline__ void wait_async0() {
    asm volatile("s_wait_asynccnt 0" ::: "memory");
}
__device__ __forceinline__ uint32_t lds_addr_of(const void* p) {
    return (uint32_t)(uintptr_t)p;
}

// Token-shift interpolation + new_state = x
__global__ __launch_bounds__(256) void rwkv_prep(
    const float* __restrict__ x, const float* __restrict__ state,
    const float* __restrict__ tmk, const float* __restrict__ tmv,
    const float* __restrict__ tmr,
    float* __restrict__ xk, float* __restrict__ xv, float* __restrict__ xr,
    float* __restrict__ out_state)
{
    int i = blockIdx.x * blockDim.x + threadIdx.x;
    if (i < CDIM) {
        float xi = x[i], si = state[i];
        float k = tmk[i], v = tmv[i], r = tmr[i];
        xk[i] = xi * k + si * (1.0f - k);
        xv[i] = xi * v + si * (1.0f - v);
        xr[i] = xi * r + si * (1.0f - r);
        out_state[i] = xi;
    }
}

// GEMV via V_WMMA_F32_16X16X4_F32, fed from LDS.
//
// Workgroup = 128 threads (4 waves) -> 64 output rows, one K-split chunk.
// K loop: 8 tiles of 64 rows x 64 cols (16 KB), double-buffered in LDS.
// Staging: thread t async-copies 16B chunks; 16 threads cover one 256B
// row segment -> fully coalesced global traffic, overlapped with WMMA of the
// previous tile via ASYNCcnt. Compute: wave w owns tile rows [16w,16w+16);
// A fragment (16x4 fp32: lanes 0-15 K={0,1}, lanes 16-31 K={2,3}) comes from
// ds_load_b64 (stride 68 floats => conflict-free). B = x segment replicated
// across all 16 N columns (broadcast ds_load_b64), so every D column holds
// the same 16 dot products; lanes 0/16 store them.
__global__ __launch_bounds__(128) void gemv_wmma_f32(
    const float* __restrict__ M, const float* __restrict__ x,
    float* __restrict__ ypart)
{
    __shared__ float tile[2][ROWS * LDS_STRIDE]; // 2 x 17408 B
    __shared__ float xlds[KCHUNK];               // 2048 B

    const int t    = threadIdx.x;
    const int lane = t & 31;
    const int wave = t >> 5;
    const int row0 = blockIdx.x * ROWS;
    const int ks   = blockIdx.y * KCHUNK;

    // stage this block's x chunk once
    *(float4*)&xlds[t * 4] = *(const float4*)(x + ks + t * 4);

    // staging geometry: 16 threads x 16B = one 256B row segment
    const int srow = t >> 4;         // 0..7
    const int scol = (t & 15) << 2;  // 0..60 step 4
    const float* gbase = M + (size_t)(row0 + srow) * CDIM + ks + scol;

    // issue tile 0
#pragma unroll
    for (int p = 0; p < 8; ++p) {
        async_copy_b128(
            lds_addr_of(&tile[0][(p * 8 + srow) * LDS_STRIDE + scol]),
            gbase + (size_t)(p * 8) * CDIM);
    }

    const int arow = wave * 16 + (lane & 15);
    const int acol = (lane >> 4) << 1; // 0 or 2: this lane's K pair
    v8f c = {};

    for (int tt = 0; tt < KCHUNK / KC; ++tt) {
        const int cur = tt & 1;
        wait_async0();       // my stage-writes for tile tt have landed
        __syncthreads();     // everyone's writes landed; prev buffer free

        if (tt + 1 < KCHUNK / KC) {
            const float* g = gbase + (tt + 1) * KC;
#pragma unroll
            for (int p = 0; p < 8; ++p) {
                async_copy_b128(
                    lds_addr_of(&tile[cur ^ 1][(p * 8 + srow) * LDS_STRIDE + scol]),
                    g + (size_t)(p * 8) * CDIM);
            }
        }

        const float* ap = &tile[cur][arow * LDS_STRIDE + acol];
        const float* bp = &xlds[tt * KC + acol];
#pragma unroll
        for (int kk = 0; kk < KC; kk += 4) {
            v2f a = *(const v2f*)(ap + kk);
            v2f b = *(const v2f*)(bp + kk);
            c = __builtin_amdgcn_wmma_f32_16x16x4_f32(
                    /*neg_a=*/false, a, /*neg_b=*/false, b,
                    /*c_mod=*/(short)0, c, /*reuse_a=*/false, /*reuse_b=*/false);
        }
    }

    // D layout: lane n (0-15) register j holds row j, column n; lane 16+n register j holds row 8+j.
    // All columns identical -> lanes 0/16 store 8 rows each.
    if ((lane & 15) == 0) {
        float* o = ypart + (size_t)blockIdx.y * CDIM + row0 + wave * 16 +
                   ((lane >> 4) << 3);
        *(float4*)(o)     = make_float4(c[0], c[1], c[2], c[3]);
        *(float4*)(o + 4) = make_float4(c[4], c[5], c[6], c[7]);
    }
}

// Reduce K-split partials, then the WKV state update with max-trick.
__global__ __launch_bounds__(256) void rwkv_wkv(
    const float* __restrict__ kkp, const float* __restrict__ vvp,
    const float* __restrict__ rrp,
    const float* __restrict__ aa, const float* __restrict__ bb,
    const float* __restrict__ pp,
    const float* __restrict__ time_first, const float* __restrict__ time_decay,
    float* __restrict__ rwkv, float* __restrict__ out)
{
    int i = blockIdx.x * blockDim.x + threadIdx.x;
    if (i >= CDIM) return;

    float kk = 0.f, vv = 0.f, rl = 0.f;
#pragma unroll
    for (int s = 0; s < KSPLIT; ++s) {
        kk += kkp[s * CDIM + i];
        vv += vvp[s * CDIM + i];
        rl += rrp[s * CDIM + i];
    }
    float r = 1.0f / (1.0f + __expf(-rl));

    float a0 = aa[i], b0 = bb[i], p0 = pp[i];

    float ww = time_first[i] + kk;
    float p  = fmaxf(p0, ww);
    float e1 = __expf(p0 - p);
    float e2 = __expf(ww - p);
    float a  = e1 * a0 + e2 * vv;
    float b  = e1 * b0 + e2;

    float ww2 = p0 + time_decay[i];
    float p2  = fmaxf(ww2, kk);
    float f1  = __expf(ww2 - p2);
    float f2  = __expf(kk - p2);

    out[2 * CDIM + i] = f1 * a0 + f2 * vv;   // new_state_a
    out[3 * CDIM + i] = f1 * b0 + f2;        // new_state_b
    out[4 * CDIM + i] = p2;                  // new_state_p

    rwkv[i] = r * (a / b);
}

// Sum K-split partials of ow@(r*wkv) into d_out[0:C).
__global__ __launch_bounds__(256) void rwkv_final(
    const float* __restrict__ outp, float* __restrict__ out)
{
    int i = blockIdx.x * blockDim.x + threadIdx.x;
    if (i >= CDIM) return;
    float o = 0.f;
#pragma unroll
    for (int s = 0; s < KSPLIT; ++s) o += outp[s * CDIM + i];
    out[i] = o;
}

extern "C" void kernel_launch(void* const* d_in, const int* in_sizes, int n_in,
                              void* d_out, int out_size, void* d_ws, size_t ws_size,
                              hipStream_t stream) {
    const float* x   = (const float*)d_in[0];
    const float* st  = (const float*)d_in[1];
    const float* sa  = (const float*)d_in[2];
    const float* sb  = (const float*)d_in[3];
    const float* sp  = (const float*)d_in[4];
    const float* tmk = (const float*)d_in[5];
    const float* tmv = (const float*)d_in[6];
    const float* tmr = (const float*)d_in[7];
    const float* tf  = (const float*)d_in[8];
    const float* td  = (const float*)d_in[9];
    const float* kw  = (const float*)d_in[10];
    const float* vw  = (const float*)d_in[11];
    const float* rw  = (const float*)d_in[12];
    const float* ow  = (const float*)d_in[13];

    float* out = (float*)d_out;
    float* ws  = (float*)d_ws;

    float* xk   = ws;
    float* xv   = ws + (size_t)CDIM;
    float* xr   = ws + (size_t)2 * CDIM;
    float* kkp  = ws + (size_t)3 * CDIM;
    float* vvp  = ws + (size_t)(3 + KSPLIT) * CDIM;       // 11C
    float* rrp  = ws + (size_t)(3 + 2 * KSPLIT) * CDIM;   // 19C
    float* rwkv = ws + (size_t)(3 + 3 * KSPLIT) * CDIM;   // 27C
    float* outp = ws + (size_t)(4 + 3 * KSPLIT) * CDIM;   // 28C

    const dim3 eg(CDIM / 256), eb(256);
    const dim3 gg(CDIM / ROWS, KSPLIT), gb(128);

    rwkv_prep<<<eg, eb, 0, stream>>>(x, st, tmk, tmv, tmr, xk, xv, xr,
                                     out + CDIM /* new_state */);

    gemv_wmma_f32<<<gg, gb, 0, stream>>>(kw, xk, kkp);
    gemv_wmma_f32<<<gg, gb, 0, stream>>>(vw, xv, vvp);
    gemv_wmma_f32<<<gg, gb, 0, stream>>>(rw, xr, rrp);

    rwkv_wkv<<<eg, eb, 0, stream>>>(kkp, vvp, rrp, sa, sb, sp, tf, td,
                                    rwkv, out);

    gemv_wmma_f32<<<gg, gb, 0, stream>>>(ow, rwkv, outp);

    rwkv_final<<<eg, eb, 0, stream>>>(outp, out);
}